// MatrixHyperlayer_56281251447198
// MI455X (gfx1250) — compile-verified
//
#include <hip/hip_runtime.h>
#include <stdint.h>

// ---------------------------------------------------------------------------
// MatrixHyperlayer for MI455X (gfx1250, wave32)
//   Kernel 1: per-row setup (threefry randoms, indices, dedup, weights)
//   Kernel 2: software-pipelined async gather (global_load_async_to_lds_b128,
//             double-buffered LDS, partial s_wait_asynccnt) + WMMA f32 16x16x4
// ---------------------------------------------------------------------------

#define IN_NUM   65536
#define OUT_NUM  65536
#define KC       4          // K gaussians per row
#define NJ       24         // k*(2+RADD+GADD) indices per row
#define EMB      128
#define LDSTR    136        // padded LDS row stride (floats): bank-conflict free
#define WAVES_PER_BLOCK 2   // kernel 2
#define RPW      8          // rows processed per wave (pipelined)

typedef float v2f __attribute__((ext_vector_type(2)));
typedef float v8f __attribute__((ext_vector_type(8)));

// ------------------------------ threefry2x32 -------------------------------
__device__ __forceinline__ unsigned rotl32(unsigned x, unsigned r) {
  return (x << r) | (x >> (32u - r));
}

__device__ __forceinline__ void threefry2x32(unsigned k0, unsigned k1,
                                             unsigned& x0, unsigned& x1) {
  const unsigned ks2 = k0 ^ k1 ^ 0x1BD11BDAu;
  x0 += k0; x1 += k1;
#define TF_R(r) { x0 += x1; x1 = rotl32(x1, r); x1 ^= x0; }
  TF_R(13u) TF_R(15u) TF_R(26u) TF_R(6u)   x0 += k1;  x1 += ks2 + 1u;
  TF_R(17u) TF_R(29u) TF_R(16u) TF_R(24u)  x0 += ks2; x1 += k0  + 2u;
  TF_R(13u) TF_R(15u) TF_R(26u) TF_R(6u)   x0 += k0;  x1 += k1  + 3u;
  TF_R(17u) TF_R(29u) TF_R(16u) TF_R(24u)  x0 += k1;  x1 += ks2 + 4u;
  TF_R(13u) TF_R(15u) TF_R(26u) TF_R(6u)   x0 += ks2; x1 += k0  + 5u;
#undef TF_R
}

// element i of jax.random.uniform(key, (524288,)) in [0,1)
__device__ __forceinline__ float jax_u01(unsigned ka, unsigned kb, unsigned i) {
  const unsigned half = (unsigned)(OUT_NUM * KC * 2) / 2u;  // 262144
  const bool hi = (i >= half);
  unsigned x0 = hi ? (i - half) : i;
  unsigned x1 = x0 + half;
  threefry2x32(ka, kb, x0, x1);
  unsigned bits = hi ? x1 : x0;
  return __uint_as_float((bits >> 9) | 0x3F800000u) - 1.0f;
}

// ------------------------- Kernel 1: row setup -----------------------------
__global__ __launch_bounds__(256)
void hl_setup(const float* __restrict__ params,
              int* __restrict__ wsIdx, float* __restrict__ wsW) {
  const int c = blockIdx.x * blockDim.x + threadIdx.x;
  if (c >= OUT_NUM) return;

  // subkeys of jax.random.split(jax.random.key(42)) -- constant-folds
  unsigned kr0 = 0u, kg0 = 2u; threefry2x32(0u, 42u, kr0, kg0);
  unsigned kr1 = 1u, kg1 = 3u; threefry2x32(0u, 42u, kr1, kg1);

  float mean[KC], sigma[KC], val[KC];
#pragma unroll
  for (int k = 0; k < KC; ++k) {
    const float* p = params + (size_t)(c * KC + k) * 3;
    const float p0 = p[0], p1 = p[1], p2 = p[2];
    mean[k] = (1.0f / (1.0f + __expf(-p0))) * (float)(IN_NUM - 1);
    const float z  = p1 + 2.0f;                                  // SIGMA_BOOST
    const float sp = fmaxf(z, 0.0f) + log1pf(__expf(-fabsf(z))); // softplus
    sigma[k] = (sp + 1e-6f) * ((float)IN_NUM * 0.2f);            // SIGMA_SCALE
    val[k]   = p2;
  }

  int ipts[NJ];
#pragma unroll
  for (int k = 0; k < KC; ++k) {
    const float m = mean[k];
    ipts[k * 6 + 0] = (int)floorf(m);
    ipts[k * 6 + 1] = (int)ceilf(m);
    const float mr = rintf(m);                 // jnp.round (ties-to-even)
    float lower = mr - 64.0f;                  // REGION*0.5
    const float upper = mr + 64.0f;
    lower = (lower < 0.0f) ? 0.0f : lower;
    lower = (upper > (float)IN_NUM) ? ((float)IN_NUM - 128.0f) : lower;
    const unsigned base = (unsigned)(c * KC + k) * 2u;
#pragma unroll
    for (int r = 0; r < 2; ++r) {
      const float u = jax_u01(kr0, kr1, base + r) * (1.0f - 1e-6f);
      ipts[k * 6 + 2 + r] = (int)floorf(u * 128.0f + lower);
      const float g = jax_u01(kg0, kg1, base + r) * (1.0f - 1e-6f) * (float)IN_NUM;
      ipts[k * 6 + 4 + r] = (int)floorf(g);
    }
  }

  // duplicate flags: stable argsort keeps first occurrence -> later dup
  bool dup[NJ];
#pragma unroll
  for (int j = 0; j < NJ; ++j) {
    bool d = false;
#pragma unroll
    for (int i = 0; i < NJ; ++i)
      if (i < j) d |= (ipts[i] == ipts[j]);
    dup[j] = d;
  }

  // per-k normalizer (dups contribute 0)
  float s[KC] = {0.0f, 0.0f, 0.0f, 0.0f};
#pragma unroll
  for (int j = 0; j < NJ; ++j) {
    if (!dup[j]) {
      const float pt = (float)ipts[j];
#pragma unroll
      for (int k = 0; k < KC; ++k) {
        const float d = pt - mean[k];
        s[k] += __expf(-0.5f * d * d / (1e-6f + sigma[k]));
      }
    }
  }

#pragma unroll
  for (int j = 0; j < NJ; ++j) {
    float v = 0.0f;
    if (!dup[j]) {
      const float pt = (float)ipts[j];
#pragma unroll
      for (int k = 0; k < KC; ++k) {
        const float d = pt - mean[k];
        v += (__expf(-0.5f * d * d / (1e-6f + sigma[k])) / s[k]) * val[k];
      }
    }
    if (ipts[j] == c) v = 0.0f;                // rows == cols -> 0
    wsIdx[(size_t)c * NJ + j] = ipts[j];
    wsW  [(size_t)c * NJ + j] = v;
  }
}

// --------------- Kernel 2: pipelined async gather + WMMA -------------------
// Issue 24 async B128 row-gathers (512B per instruction) for the NEXT row
// into the other LDS buffer, wait s_wait_asynccnt 24 (async ops complete
// in order => previous row's 24 transfers are done), then consume the
// current buffer with 6x8 V_WMMA_F32_16X16X4_F32.

__device__ __forceinline__ void hl_issue_row(const float* __restrict__ x,
                                             const int* __restrict__ idxr,
                                             unsigned ldsDstBase, int lane) {
  const unsigned long long xBase = (unsigned long long)x;
#pragma unroll
  for (int j = 0; j < NJ; ++j) {
    const int ij = idxr[j];
    const unsigned voff = (unsigned)ij * (EMB * 4u) + (unsigned)lane * 16u;
    const unsigned dst  = ldsDstBase + (unsigned)j * (LDSTR * 4u) + (unsigned)lane * 16u;
    asm volatile("global_load_async_to_lds_b128 %0, %1, %2"
                 :: "v"(dst), "v"(voff), "s"(xBase)
                 : "memory");
  }
}

__device__ __forceinline__ void hl_compute_row(const float* __restrict__ G,
                                               const float* __restrict__ wr,
                                               float* __restrict__ outRow,
                                               int lane) {
  // A[m,k] = w[4g+k] for all m. ISA layout: lanes 0-15 hold (K=0,K=1),
  // lanes 16-31 hold (K=2,K=3) in (VGPR0,VGPR1).
  float a0[6], a1[6];
#pragma unroll
  for (int g = 0; g < 6; ++g) {
    const float w0 = wr[4 * g + 0], w1 = wr[4 * g + 1];
    const float w2 = wr[4 * g + 2], w3 = wr[4 * g + 3];
    a0[g] = (lane < 16) ? w0 : w2;
    a1[g] = (lane < 16) ? w1 : w3;
  }

  v8f acc[8];
#pragma unroll
  for (int t = 0; t < 8; ++t) acc[t] = (v8f){0.f, 0.f, 0.f, 0.f, 0.f, 0.f, 0.f, 0.f};

  const int col   = lane & 15;
  const int rbase = (lane < 16) ? 0 : 2;  // B: K=0/2 lane halves (VGPR0), K=1/3 (VGPR1)
#pragma unroll
  for (int g = 0; g < 6; ++g) {
    v2f A; A.x = a0[g]; A.y = a1[g];
    const float* brow = G + (4 * g + rbase) * LDSTR;
#pragma unroll
    for (int t = 0; t < 8; ++t) {
      v2f B;
      B.x = brow[t * 16 + col];            // ds_load: B[k=0/2, n]
      B.y = brow[LDSTR + t * 16 + col];    // ds_load: B[k=1/3, n]
      acc[t] = __builtin_amdgcn_wmma_f32_16x16x4_f32(
          false, A, false, B, (short)0, acc[t], false, false);
    }
  }

  // row M=0 of C (element 0 of fragment, lanes 0-15) is the result
  if (lane < 16) {
#pragma unroll
    for (int t = 0; t < 8; ++t)
      outRow[t * 16 + lane] = acc[t][0];
  }
}

__global__ __launch_bounds__(WAVES_PER_BLOCK * 32)
void hl_gather_wmma(const float* __restrict__ x,
                    const int* __restrict__ wsIdx,
                    const float* __restrict__ wsW,
                    float* __restrict__ out) {
  // 2 waves * 2 buffers * 24 * 136 floats = 51 KB
  __shared__ float sh[WAVES_PER_BLOCK * 2 * NJ * LDSTR];

  const int lane    = threadIdx.x & 31;
  const int wv      = threadIdx.x >> 5;
  const int rowBase = (blockIdx.x * WAVES_PER_BLOCK + wv) * RPW;

  float* buf[2] = { &sh[(wv * 2 + 0) * NJ * LDSTR],
                    &sh[(wv * 2 + 1) * NJ * LDSTR] };
  const unsigned lbase[2] = { (unsigned)(uintptr_t)buf[0],
                              (unsigned)(uintptr_t)buf[1] };

  // prologue: stage row 0
  hl_issue_row(x, wsIdx + (size_t)rowBase * NJ, lbase[0], lane);

#pragma unroll
  for (int i = 0; i < RPW - 1; ++i) {
    // stage row i+1 into the other buffer while row i is in flight/consumed
    hl_issue_row(x, wsIdx + (size_t)(rowBase + i + 1) * NJ, lbase[(i + 1) & 1], lane);
    // async ops complete in order: <=24 outstanding => row i fully in LDS
    asm volatile("s_wait_asynccnt 24" ::: "memory");
    hl_compute_row(buf[i & 1], wsW + (size_t)(rowBase + i) * NJ,
                   out + (size_t)(rowBase + i) * EMB, lane);
  }

  // epilogue: last row needs a full drain
  asm volatile("s_wait_asynccnt 0" ::: "memory");
  hl_compute_row(buf[(RPW - 1) & 1], wsW + (size_t)(rowBase + RPW - 1) * NJ,
                 out + (size_t)(rowBase + RPW - 1) * EMB, lane);
}

// ------------------------------- launcher ----------------------------------
extern "C" void kernel_launch(void* const* d_in, const int* in_sizes, int n_in,
                              void* d_out, int out_size, void* d_ws, size_t ws_size,
                              hipStream_t stream) {
  (void)in_sizes; (void)n_in; (void)out_size; (void)ws_size;
  const float* params = (const float*)d_in[0];   // (K*OUT_NUM, 3) f32
  const float* x      = (const float*)d_in[1];   // (IN_NUM, EMB) f32
  float* out = (float*)d_out;                    // (OUT_NUM, EMB) f32

  int*   wsIdx = (int*)d_ws;
  float* wsW   = (float*)((char*)d_ws + (size_t)OUT_NUM * NJ * sizeof(int));

  hl_setup<<<OUT_NUM / 256, 256, 0, stream>>>(params, wsIdx, wsW);

  const int rowsPerBlock = WAVES_PER_BLOCK * RPW;   // 16
  hl_gather_wmma<<<OUT_NUM / rowsPerBlock, WAVES_PER_BLOCK * 32, 0, stream>>>(
      x, wsIdx, wsW, out);
}